// PeepholeLSTM_17463337026010
// MI455X (gfx1250) — compile-verified
//
#include <hip/hip_runtime.h>

typedef __attribute__((ext_vector_type(2))) float v2f;
typedef __attribute__((ext_vector_type(8))) float v8f;

#define SEQ   512
#define BATCH 64
#define INP   512
#define HS    1024
#define GH    4096            // 4*HS
#define KTOT  1536            // INP + HS
#define KP    64              // K panel staged in LDS
#define NPANEL (KTOT / KP)    // 24
#define LDP   66              // padded LDS row stride (floats) to dodge bank conflicts
#define NT    16              // n-columns (of HS) per block
#define MT    32              // batch rows per block
#define NCT   (GH / 16)       // 256 column tiles of W
#define KSTEPS (KTOT / 4)     // 384 wmma k-steps
#define WP_ELEMS (NCT * KSTEPS * 64)   // repacked W, 6.29M floats = 24 MB

// ---------------------------------------------------------------------------
// Prep: repack W (1536 x 4096, row major) into WMMA B-fragment stream order:
//   Wp[ct][s][lane][p] = W[4*s + 2*(lane>>4) + p][16*ct + (lane&15)]
// (B 4x16 f32 fragment: v0/v1 per lane; lanes 0-15 hold K=0,1; lanes 16-31 K=2,3)
// Also zeros the c-state so every kernel_launch call starts from c0 = 0.
// ---------------------------------------------------------------------------
__global__ __launch_bounds__(256) void lstm_prep(const float* __restrict__ W,
                                                 float* __restrict__ wp,
                                                 float* __restrict__ c_ws) {
    int idx = blockIdx.x * 256 + threadIdx.x;
    if (idx < WP_ELEMS) {
        int p   = idx & 1;
        int L   = (idx >> 1) & 31;
        int tmp = idx >> 6;              // ct*KSTEPS + s
        int s   = tmp % KSTEPS;
        int ct  = tmp / KSTEPS;
        int k   = 4 * s + 2 * (L >> 4) + p;
        int n   = 16 * ct + (L & 15);
        wp[idx] = W[(size_t)k * GH + n];
    }
    if (idx < BATCH * HS) c_ws[idx] = 0.0f;
}

// ---------------------------------------------------------------------------
// One fused LSTM timestep.
//  grid = (HS/NT, BATCH/MT) = (64, 2), block = 256 threads = 8 waves.
//  wave w: mt = w&1 (16-row M tile inside 32-row chunk), g = w>>1 (gate 0..3).
//  Each wave: one 16x16 fp32 tile, K=1536 via v_wmma_f32_16x16x4_f32.
//  Gates exchanged through LDS, then peephole cell update fused in-block.
// ---------------------------------------------------------------------------
__global__ __launch_bounds__(256) void lstm_step(
    const float* __restrict__ x_t,     // [64, 512]
    const float* __restrict__ h_prev,  // [64, 1024], nullptr at t==0
    const float* __restrict__ wp,      // repacked W
    const float* __restrict__ bias,    // [4096]
    const float* __restrict__ peep_i,  // [1024]
    const float* __restrict__ peep_f,
    const float* __restrict__ peep_o,
    float* __restrict__ c_ws,          // [64, 1024] cell state
    float* __restrict__ h_out,         // [64, 1024] = hidden_seq[t]
    float* __restrict__ cT,            // non-null only at t == SEQ-1
    float* __restrict__ hT)
{
    __shared__ __align__(16) float lds_a[MT * LDP];   // 32 x 64 A panel (padded)
    __shared__ __align__(16) float lds_g[4 * MT * NT]; // 4 gates x 32 x 16

    const int tid    = threadIdx.x;
    const int lane   = tid & 31;
    const int wave   = tid >> 5;
    const int mt     = wave & 1;
    const int g      = wave >> 1;
    const int nchunk = blockIdx.x;     // 0..63
    const int mchunk = blockIdx.y;     // 0..1

    const int ct = g * (HS / 16) / 1 * 1;            // gate base tile ...
    const int coltile = g * 64 + nchunk;             // global W column tile 0..255
    const float* wps = wp + (size_t)coltile * (KSTEPS * 64);
    (void)ct;

    v8f acc0 = {};
    v8f acc1 = {};

    // A-fragment addressing (16x4 f32 A layout):
    //   lane L: row = L%16 ; cols = 2*(L/16) + {0,1}
    const int arow  = mt * 16 + (lane & 15);
    const int acol0 = 2 * (lane >> 4);

    for (int pa = 0; pa < NPANEL; ++pa) {
        const int k0 = pa * KP;
        // --- cooperative stage of 32x64 A panel ([x_t | h_prev]) into LDS ---
        {
            const int row = tid >> 3;            // 0..31
            const int kk0 = (tid & 7) * 8;       // 0,8,...,56
            const int m   = mchunk * MT + row;
            float* dst = &lds_a[row * LDP + kk0];
            if (k0 < INP) {                      // panel fully in x (uniform)
                const float* src = x_t + (size_t)m * INP + k0 + kk0;
                #pragma unroll
                for (int i = 0; i < 8; ++i) dst[i] = src[i];
            } else if (h_prev) {                 // panel fully in h (uniform)
                const float* src = h_prev + (size_t)m * HS + (k0 - INP) + kk0;
                #pragma unroll
                for (int i = 0; i < 8; ++i) dst[i] = src[i];
            } else {                             // t == 0: h0 = 0
                #pragma unroll
                for (int i = 0; i < 8; ++i) dst[i] = 0.0f;
            }
        }
        __syncthreads();

        // --- 16 wmma k-steps over the panel, 2 accumulator chains for ILP ---
        const float* wpp = wps + (size_t)(pa * 16) * 64 + lane * 2;
        __builtin_prefetch(wpp + 16 * 64, 0, 1);   // prefetch next panel's B stream
        #pragma unroll
        for (int ks = 0; ks < 16; ks += 2) {
            v2f a0 = *(const v2f*)&lds_a[arow * LDP + ks * 4 + acol0];
            v2f b0 = *(const v2f*)(wpp + (size_t)ks * 64);
            acc0 = __builtin_amdgcn_wmma_f32_16x16x4_f32(
                false, a0, false, b0, (short)0, acc0, false, false);
            v2f a1 = *(const v2f*)&lds_a[arow * LDP + (ks + 1) * 4 + acol0];
            v2f b1 = *(const v2f*)(wpp + (size_t)(ks + 1) * 64);
            acc1 = __builtin_amdgcn_wmma_f32_16x16x4_f32(
                false, a1, false, b1, (short)0, acc1, false, false);
        }
        __syncthreads();
    }

    // --- spill gate tiles to LDS (C/D layout: VGPR r, lane L -> M=r+8*(L/16), N=L%16)
    {
        const int hi = lane >> 4, lo = lane & 15;
        #pragma unroll
        for (int r = 0; r < 8; ++r) {
            int ml = mt * 16 + r + 8 * hi;
            lds_g[(g * MT + ml) * NT + lo] = acc0[r] + acc1[r];
        }
    }
    __syncthreads();

    // --- fused peephole LSTM cell update: 512 (b,n) pairs, 2 per thread ---
    #pragma unroll
    for (int it = 0; it < 2; ++it) {
        int q  = tid + 256 * it;
        int ml = q >> 4;                 // 0..31
        int nl = q & 15;
        int b  = mchunk * MT + ml;
        int n  = nchunk * NT + nl;

        float gi = lds_g[(0 * MT + ml) * NT + nl] + bias[n];
        float gj = lds_g[(1 * MT + ml) * NT + nl] + bias[HS + n];
        float gf = lds_g[(2 * MT + ml) * NT + nl] + bias[2 * HS + n];
        float go = lds_g[(3 * MT + ml) * NT + nl] + bias[3 * HS + n];

        float cp = h_prev ? c_ws[(size_t)b * HS + n] : 0.0f;
        float fs = 1.0f / (1.0f + __expf(-(gf + cp * peep_f[n])));
        float is = 1.0f / (1.0f + __expf(-(gi + cp * peep_i[n])));
        float cn = fs * cp + is * tanhf(gj);
        float os = 1.0f / (1.0f + __expf(-(go + cn * peep_o[n])));
        float hn = os * tanhf(cn);

        c_ws[(size_t)b * HS + n]  = cn;
        h_out[(size_t)b * HS + n] = hn;
        if (cT) {
            cT[(size_t)b * HS + n] = cn;
            hT[(size_t)b * HS + n] = hn;
        }
    }
}

// ---------------------------------------------------------------------------
extern "C" void kernel_launch(void* const* d_in, const int* in_sizes, int n_in,
                              void* d_out, int out_size, void* d_ws, size_t ws_size,
                              hipStream_t stream) {
    const float* x      = (const float*)d_in[0];
    const float* W      = (const float*)d_in[1];
    const float* bias   = (const float*)d_in[2];
    const float* peep_i = (const float*)d_in[3];
    const float* peep_f = (const float*)d_in[4];
    const float* peep_o = (const float*)d_in[5];
    float* out = (float*)d_out;

    float* wp   = (float*)d_ws;          // 24 MB repacked W
    float* c_ws = wp + WP_ELEMS;         // 256 KB cell state

    lstm_prep<<<(WP_ELEMS + 255) / 256, 256, 0, stream>>>(W, wp, c_ws);

    const size_t HSEQ = (size_t)SEQ * BATCH * HS;
    dim3 grid(HS / NT, BATCH / MT);      // (64, 2)
    for (int t = 0; t < SEQ; ++t) {
        const float* x_t    = x + (size_t)t * BATCH * INP;
        const float* h_prev = t ? (out + (size_t)(t - 1) * BATCH * HS) : nullptr;
        float* h_out = out + (size_t)t * BATCH * HS;
        float* cTp = (t == SEQ - 1) ? (out + HSEQ) : nullptr;
        float* hTp = (t == SEQ - 1) ? (out + HSEQ + (size_t)BATCH * HS) : nullptr;
        lstm_step<<<grid, 256, 0, stream>>>(x_t, h_prev, wp, bias,
                                            peep_i, peep_f, peep_o,
                                            c_ws, h_out, cTp, hTp);
    }
}